// SparseEdgeEmbeddingV2_61048665145757
// MI455X (gfx1250) — compile-verified
//
#include <hip/hip_runtime.h>
#include <hip/hip_bf16.h>
#include <math.h>

typedef __attribute__((ext_vector_type(2))) float v2f;
typedef __attribute__((ext_vector_type(8))) float v8f;

#define NPTS   16384
#define KNN    32
#define NSIG   16
#define RPB    16      // rows per block
#define CHUNK  256     // columns staged per iteration
#define LISTLEN 33     // K+1 (self included, dropped at the end)

__device__ __forceinline__ void insert_sorted(float* bd, int* bi, float d, int idx) {
    // sorted ascending list of LISTLEN entries; lane 0 of one wave owns each row
    if (d >= bd[LISTLEN - 1]) return;
    int p = LISTLEN - 1;
    while (p > 0 && bd[p - 1] > d) {
        bd[p] = bd[p - 1];
        bi[p] = bi[p - 1];
        --p;
    }
    bd[p] = d;
    bi[p] = idx;
}

__global__ __launch_bounds__(256) void knn_rbf_wmma_kernel(
        const float* __restrict__ coord,   // [NPTS][3]
        float* __restrict__ outW,          // [NPTS*KNN][NSIG]
        int* __restrict__ outIdx)          // [NPTS*KNN][2]
{
    __shared__ float sRx[RPB], sRy[RPB], sRz[RPB], sRsq[RPB];
    __shared__ float sCx[CHUNK], sCy[CHUNK], sCz[CHUNK], sCsq[CHUNK];
    __shared__ float sDist[RPB * CHUNK];          // 16 KB: squared distances
    __shared__ float sBestD[RPB * LISTLEN];       // sorted ascending d^2
    __shared__ int   sBestI[RPB * LISTLEN];

    const int tid  = threadIdx.x;
    const int wv   = tid >> 5;      // wave id 0..7 (wave32)
    const int lane = tid & 31;
    const int half = lane >> 4;     // 0: lanes 0-15, 1: lanes 16-31
    const int hl   = lane & 15;
    const int row0 = blockIdx.x * RPB;

    // ---- load this block's 16 row points, init top-K lists ----
    if (tid < RPB) {
        float x = coord[(row0 + tid) * 3 + 0];
        float y = coord[(row0 + tid) * 3 + 1];
        float z = coord[(row0 + tid) * 3 + 2];
        sRx[tid] = x; sRy[tid] = y; sRz[tid] = z;
        sRsq[tid] = x * x + y * y + z * z;
    }
    for (int s = tid; s < RPB * LISTLEN; s += 256) {
        sBestD[s] = 3.0e38f;
        sBestI[s] = 0;
    }
    __syncthreads();

    // ---- A fragment (16x4 f32), branch-free:
    //      lanes 0-15 -> (K0,K1)=(x,y) ; lanes 16-31 -> (K2,K3)=(z,0)
    v2f afrag;
    {
        const float ax = sRx[hl], ay = sRy[hl], az = sRz[hl];
        afrag.x = half ? az : ax;
        afrag.y = half ? 0.0f : ay;
    }
    float rsqv[8];
#pragma unroll
    for (int v = 0; v < 8; ++v) rsqv[v] = sRsq[v + 8 * half];

    // ---- stream over all columns in chunks of 256 ----
    for (int base = 0; base < NPTS; base += CHUNK) {
        {   // stage column coords + squared norms
            int g = base + tid;
            float x = coord[g * 3 + 0];
            float y = coord[g * 3 + 1];
            float z = coord[g * 3 + 2];
            sCx[tid] = x; sCy[tid] = y; sCz[tid] = z;
            sCsq[tid] = x * x + y * y + z * z;
        }
        if (base + CHUNK < NPTS)
            __builtin_prefetch(&coord[(base + CHUNK + tid) * 3], 0, 0);
        __syncthreads();

        // each wave computes two 16x16 tiles: columns [wv*16) and [(wv+8)*16)
#pragma unroll
        for (int t = 0; t < 2; ++t) {
            const int tb = (wv + t * 8) * 16;
            v2f bfrag;   // B (4x16 f32), branch-free selects (no exec branching)
            {
                const float bx = sCx[tb + hl], by = sCy[tb + hl], bz = sCz[tb + hl];
                bfrag.x = half ? bz : bx;
                bfrag.y = half ? 0.0f : by;
            }

            v8f acc = {};
            acc = __builtin_amdgcn_wmma_f32_16x16x4_f32(
                      /*neg_a=*/false, afrag, /*neg_b=*/false, bfrag,
                      /*c_mod=*/(short)0, acc, /*reuse_a=*/false, /*reuse_b=*/false);

            const int col = tb + hl;           // this lane's column in the chunk
            const float csq = sCsq[col];
#pragma unroll
            for (int v = 0; v < 8; ++v) {      // D element: M = v + 8*half, N = col
                const int m = v + 8 * half;
                // store d^2 only; sqrt is monotonic so selection runs on d^2
                sDist[m * CHUNK + col] = rsqv[v] + csq - 2.0f * acc[v];
            }
        }
        __syncthreads();

        // ---- wave-parallel top-K on d^2: wave wv owns rows 2wv and 2wv+1 ----
        for (int rr = 0; rr < 2; ++rr) {
            const int r = wv * 2 + rr;
            float* bd = &sBestD[r * LISTLEN];
            int*   bi = &sBestI[r * LISTLEN];
            float w0 = (lane == 0) ? bd[LISTLEN - 1] : 0.0f;
            float worst = __shfl(w0, 0, 32);
            for (int j0 = 0; j0 < CHUNK; j0 += 32) {
                const int j = j0 + lane;
                const float dv = sDist[r * CHUNK + j];
                unsigned mask = (unsigned)__ballot(dv < worst);
                while (mask) {
                    const int src = __ffs(mask) - 1;
                    mask &= mask - 1;
                    const float dc = __shfl(dv, src, 32);
                    const int   jc = __shfl(j,  src, 32);
                    if (lane == 0) insert_sorted(bd, bi, dc, base + jc);
                    w0 = (lane == 0) ? bd[LISTLEN - 1] : 0.0f;
                    worst = __shfl(w0, 0, 32);
                }
            }
        }
        __syncthreads();
    }

    // ---- epilogue: Gaussian RBF over 16 sigma channels + index pairs ----
    float inv2s2[NSIG];
#pragma unroll
    for (int s = 0; s < NSIG; ++s) {
        const float sg = 0.1f + (4.9f / 15.0f) * (float)s;   // linspace(0.1, 5.0, 16)
        inv2s2[s] = 1.0f / (2.0f * sg * sg);
    }

    for (int pr = tid; pr < RPB * KNN; pr += 256) {
        const int r = pr >> 5;                 // local row
        const int k = pr & 31;                 // neighbor slot (0..31)
        const float d2raw = sBestD[r * LISTLEN + 1 + k];  // drop entry 0 (self)
        const int   nidx  = sBestI[r * LISTLEN + 1 + k];
        const int   row   = row0 + r;
        const long long e = (long long)row * KNN + k;
        // mirror reference rounding: dist = sqrt(max(d2,0)); then use dist^2
        const float dist = sqrtf(fmaxf(d2raw, 0.0f));
        const float d2 = dist * dist;
        float* wp = outW + e * NSIG;
#pragma unroll
        for (int s = 0; s < NSIG; s += 4) {
            float4 v;
            v.x = __expf(-d2 * inv2s2[s + 0]);
            v.y = __expf(-d2 * inv2s2[s + 1]);
            v.z = __expf(-d2 * inv2s2[s + 2]);
            v.w = __expf(-d2 * inv2s2[s + 3]);
            v.x = (v.x == v.x) ? v.x : 0.0f;   // NaN guard per reference
            v.y = (v.y == v.y) ? v.y : 0.0f;
            v.z = (v.z == v.z) ? v.z : 0.0f;
            v.w = (v.w == v.w) ? v.w : 0.0f;
            *(float4*)(wp + s) = v;
        }
        outIdx[2 * e + 0] = row;
        outIdx[2 * e + 1] = nidx;
    }
}

extern "C" void kernel_launch(void* const* d_in, const int* in_sizes, int n_in,
                              void* d_out, int out_size, void* d_ws, size_t ws_size,
                              hipStream_t stream) {
    (void)in_sizes; (void)n_in; (void)out_size; (void)d_ws; (void)ws_size;
    const float* coord = (const float*)d_in[0];
    // d_out layout: [1, N*K, NSIG] f32 weights, then [N*K, 2] i32 index pairs
    float* outW   = (float*)d_out;
    int*   outIdx = (int*)((float*)d_out + (size_t)NPTS * KNN * NSIG);
    dim3 grid(NPTS / RPB);   // 1024 blocks x 256 threads (8 wave32s)
    knn_rbf_wmma_kernel<<<grid, 256, 0, stream>>>(coord, outW, outIdx);
}